// Conv_23665269801519
// MI455X (gfx1250) — compile-verified
//
#include <hip/hip_runtime.h>
#include <hip/hip_bf16.h>

// CDNA5 / gfx1250, wave32.
//
// Math: per 8x16 image,  Y = rowpad( vsum3( X * H ) ),  where
//   H[j,c] = s0 if j==c-1, s1 if j==c, s2 if j==c+1 (for 1<=c<=14), else 0
//   s[k]   = sum of row k of the 3x3 weight matrix.
// X*H is done with V_WMMA_F32_16X16X4_F32: two images stacked vertically form
// a 16x16 A-matrix, K=16 accumulated over 4 chained WMMAs against the banded
// 16x16 H held in registers (B operand). The WMMA D layout places image-a rows
// in lanes 0-15 and image-b rows in lanes 16-31, one row per acc VGPR, so the
// vertical 3-row sum is pure in-lane VALU. Borders: H zeroes cols 0/15; rows
// 0/7 are stored as literal 0.

typedef __attribute__((ext_vector_type(2))) float v2f;
typedef __attribute__((ext_vector_type(8))) float v8f;

__global__ __launch_bounds__(256) void conv3x3_wmma_f32(
    const float* __restrict__ x,   // (N_img, 128) f32
    const float* __restrict__ w,   // (3,3) f32
    float* __restrict__ out,       // (N_img, 128) f32
    int n_pairs)                   // N_img / 2
{
    // Uniform per-row kernel sums s[k] = W[k,0]+W[k,1]+W[k,2] (scalar loads).
    const float s0 = w[0] + w[1] + w[2];
    const float s1 = w[3] + w[4] + w[5];
    const float s2 = w[6] + w[7] + w[8];

    const int lane = threadIdx.x & 31;
    const int n    = lane & 15;   // A: row M this lane feeds; B/D: column N
    const int h    = lane >> 4;   // lane half (selects K+2 for A/B, M+8 for D)

    // ---- Build B operand = banded H (16x16) in WMMA B layout --------------
    // Chunk c covers K rows 4c..4c+3. Component v of the v2f holds row
    // (4c + v + 2h), column n  (mirrors the documented f32 A-matrix layout).
    v2f B[4];
#pragma unroll
    for (int c = 0; c < 4; ++c) {
#pragma unroll
        for (int v = 0; v < 2; ++v) {
            const int j = 4 * c + 2 * h + v;
            float hv = 0.0f;
            if (n >= 1 && n <= 14) {
                if (j == n - 1)      hv = s0;
                else if (j == n)     hv = s1;
                else if (j == n + 1) hv = s2;
            }
            B[c][v] = hv;
        }
    }

    const int waves_per_blk = blockDim.x >> 5;
    const int wave   = blockIdx.x * waves_per_blk + (threadIdx.x >> 5);
    const int stride = gridDim.x * waves_per_blk;

    for (int p = wave; p < n_pairs; p += stride) {
        const float* xb = x + (size_t)p * 256;   // two contiguous images (1 KB)

        // Prefetch the next pair this wave will touch (global_prefetch_b8).
        if (p + stride < n_pairs)
            __builtin_prefetch(x + (size_t)(p + stride) * 256 + n * 16, 0, 0);

        // ---- A operand: X2 (16x16) in native WMMA f32 A layout ------------
        // lane (either half) = row M = n; VGPR pair c holds K = 4c+2h+{0,1}.
        // float2 loads, 8B aligned, whole wave covers the contiguous 1 KB.
        const v2f A0 = *(const v2f*)(xb + n * 16 + 0 * 4 + h * 2);
        const v2f A1 = *(const v2f*)(xb + n * 16 + 1 * 4 + h * 2);
        const v2f A2 = *(const v2f*)(xb + n * 16 + 2 * 4 + h * 2);
        const v2f A3 = *(const v2f*)(xb + n * 16 + 3 * 4 + h * 2);

        // ---- Z = X2 * H : 4 chained v_wmma_f32_16x16x4_f32 ----------------
        v8f acc = {0.f, 0.f, 0.f, 0.f, 0.f, 0.f, 0.f, 0.f};
        acc = __builtin_amdgcn_wmma_f32_16x16x4_f32(false, A0, false, B[0],
                                                    (short)0, acc, false, false);
        acc = __builtin_amdgcn_wmma_f32_16x16x4_f32(false, A1, false, B[1],
                                                    (short)0, acc, false, false);
        acc = __builtin_amdgcn_wmma_f32_16x16x4_f32(false, A2, false, B[2],
                                                    (short)0, acc, false, false);
        acc = __builtin_amdgcn_wmma_f32_16x16x4_f32(false, A3, false, B[3],
                                                    (short)0, acc, false, false);

        // ---- Vertical 3-row sum + zero border rows, store -----------------
        // acc[v]: lanes 0-15 = Z[v] of image a, lanes 16-31 = Z[v] of image b.
        const float z0 = acc[0], z1 = acc[1], z2 = acc[2], z3 = acc[3];
        const float z4 = acc[4], z5 = acc[5], z6 = acc[6], z7 = acc[7];

        float* ob = out + (size_t)p * 256 + h * 128 + n;
        ob[0 * 16] = 0.0f;            // top border row
        ob[1 * 16] = z0 + z1 + z2;
        ob[2 * 16] = z1 + z2 + z3;
        ob[3 * 16] = z2 + z3 + z4;
        ob[4 * 16] = z3 + z4 + z5;
        ob[5 * 16] = z4 + z5 + z6;
        ob[6 * 16] = z5 + z6 + z7;
        ob[7 * 16] = 0.0f;            // bottom border row
    }
}

extern "C" void kernel_launch(void* const* d_in, const int* in_sizes, int n_in,
                              void* d_out, int out_size, void* d_ws, size_t ws_size,
                              hipStream_t stream) {
    const float* x = (const float*)d_in[0];
    const float* w = (const float*)d_in[1];
    float* out = (float*)d_out;

    const int n_img   = in_sizes[0] / 128;   // 524288
    const int n_pairs = n_img / 2;           // 262144 (even by construction)

    const int threads = 256;                 // 8 wave32 waves per block
    const int blocks  = 4096;                // grid-stride: 8 pairs per wave
    hipLaunchKernelGGL(conv3x3_wmma_f32, dim3(blocks), dim3(threads), 0, stream,
                       x, w, out, n_pairs);
}